// TemporalGraphEncoder_11596411699517
// MI455X (gfx1250) — compile-verified
//
#include <hip/hip_runtime.h>
#include <math.h>

// ---------------------------------------------------------------------------
// Problem constants (from reference): N=50000, E=1600000, D=128, H=4, C=32
// ---------------------------------------------------------------------------
#define NN 50000
#define EE 1600000LL
#define DD 128
#define HH 4
#define CC 32

typedef __attribute__((ext_vector_type(2))) float f32x2;
typedef __attribute__((ext_vector_type(8))) float f32x8;

__device__ __forceinline__ float geluf(float x) {
  // exact GELU: x * 0.5 * (1 + erf(x/sqrt(2)))
  return 0.5f * x * (1.0f + erff(x * 0.70710678118654752440f));
}

__device__ __forceinline__ void atomicMaxF(float* addr, float v) {
  // monotonic int encoding trick; init value must be -inf (0xff800000)
  if (v >= 0.0f) atomicMax((int*)addr, __float_as_int(v));
  else           atomicMin((unsigned int*)addr, __float_as_uint(v));
}

// ---------------------------------------------------------------------------
// Spectral normalization: power iteration on a 128x128 matrix.
// grid.x = 2 (block 0: W_in, block 1: W_ode), 128 threads.
// ---------------------------------------------------------------------------
__global__ void spectral_kernel(const float* __restrict__ Wa, const float* __restrict__ Wb,
                                float* __restrict__ Oa, float* __restrict__ Ob) {
  const float* W = (blockIdx.x == 0) ? Wa : Wb;
  float* O       = (blockIdx.x == 0) ? Oa : Ob;
  __shared__ float u[DD], v[DD], red[DD];
  const int t = threadIdx.x;
  u[t] = 0.08838834764831845f;  // 1/sqrt(128)
  __syncthreads();
  float sigma = 0.0f;
  for (int it = 0; it < 5; ++it) {
    // v = W^T u ; normalize
    float a = 0.0f;
    for (int i = 0; i < DD; ++i) a += W[i * DD + t] * u[i];
    red[t] = a * a; __syncthreads();
    for (int s = 64; s > 0; s >>= 1) { if (t < s) red[t] += red[t + s]; __syncthreads(); }
    const float nv = sqrtf(red[0]); __syncthreads();
    v[t] = a / (nv + 1e-12f); __syncthreads();
    // u = W v ; normalize ; sigma = u . (W v) = |Wv|^2 / (|Wv| + eps)
    float b = 0.0f;
    for (int j = 0; j < DD; ++j) b += W[t * DD + j] * v[j];
    red[t] = b * b; __syncthreads();
    for (int s = 64; s > 0; s >>= 1) { if (t < s) red[t] += red[t + s]; __syncthreads(); }
    const float nu = sqrtf(red[0]); __syncthreads();
    sigma = (nu * nu) / (nu + 1e-12f);
    u[t] = b / (nu + 1e-12f); __syncthreads();
  }
  const float inv = 1.0f / (sigma + 1e-12f);
  for (int i = t; i < DD * DD; i += DD) O[i] = W[i] * inv;
}

// ---------------------------------------------------------------------------
// WMMA f32 GEMM: C[n,128] = act(A[n,128] @ W[128,128] + bias)
// 256 threads = 8 waves; block covers 128 rows; wave w -> rows [16w,16w+16).
// Uses V_WMMA_F32_16X16X4_F32 (full f32 precision on the matrix pipe).
//
// LDS holds A and W interleaved by K-PAIRS so each WMMA operand is one
// contiguous 8B element -> single conflict-free ds_load_b64 into an even
// VGPR pair (no repack movs):
//   Al[k2*288 + row*2 + (k&1)]   k2 = k>>1, row in [0,128)
//   Wl[k2*288 + col*2 + (k&1)]
// Stride 288 == 32 (mod 64 banks): the two half-wave K rows (hi=0/1) hit
// complementary 32-bank sets -> conflict-free.
// All 8 B fragments are fetched before the WMMA burst so the compiler can
// use partial s_wait_dscnt staging instead of a full drain every 2 WMMAs.
// ---------------------------------------------------------------------------
#define P_STR 288
#define GEMM_LDS_BYTES ((size_t)(64 * P_STR * 2) * sizeof(float))

__global__ __launch_bounds__(256) void gemm128_kernel(
    const float* __restrict__ A, const float* __restrict__ W,
    const float* __restrict__ bias, float* __restrict__ Cout,
    int nrows, int act) {
  extern __shared__ float lds[];
  float* Al = lds;                 // 64 * 288
  float* Wl = lds + 64 * P_STR;    // 64 * 288
  const int tid = threadIdx.x;
  const int brow = blockIdx.x * 128;

  for (int i = tid; i < DD * 32; i += 256) {       // 128 rows x 32 float4
    const int r = i >> 5, c4 = (i & 31) << 2;
    // W[r][c4..c4+3]: k = r -> pair (r>>1, r&1), columns interleave *2
    const float4 w = *(const float4*)(W + r * DD + c4);
    float* wp = Wl + (r >> 1) * P_STR + c4 * 2 + (r & 1);
    wp[0] = w.x; wp[2] = w.y; wp[4] = w.z; wp[6] = w.w;
    // A[gr][c4..c4+3]: k = c4+j -> pair ((c4+j)>>1, (c4+j)&1), row interleave *2
    const int gr = brow + r;
    float4 a4 = make_float4(0.f, 0.f, 0.f, 0.f);
    if (gr < nrows) a4 = *(const float4*)(A + (size_t)gr * DD + c4);
    float* ap = Al + (c4 >> 1) * P_STR + r * 2;    // c4 is even
    ap[0]         = a4.x;   // k = c4   -> pair c4>>1, p=0
    ap[1]         = a4.y;   // k = c4+1 -> pair c4>>1, p=1
    ap[P_STR]     = a4.z;   // k = c4+2 -> pair c4>>1 + 1, p=0
    ap[P_STR + 1] = a4.w;   // k = c4+3
  }
  __syncthreads();

  const int wave = tid >> 5, lane = tid & 31;
  const int lr = lane & 15, hi = lane >> 4;
  const int mrow = wave * 16;

  f32x8 acc[8];
  const f32x8 z = {0.f, 0.f, 0.f, 0.f, 0.f, 0.f, 0.f, 0.f};
#pragma unroll
  for (int t = 0; t < 8; ++t) acc[t] = z;

  const float* Arow = Al + (mrow + lr) * 2;   // + k2*P_STR
  const float* Wcol = Wl + lr * 2;            // + k2*P_STR + t*32

  for (int kk = 0; kk < 32; ++kk) {
    const int k2 = 2 * kk + hi;               // K-pair index for this half-wave
    const f32x2 a = *(const f32x2*)(Arow + k2 * P_STR);
    f32x2 b[8];
#pragma unroll
    for (int t = 0; t < 8; ++t) b[t] = *(const f32x2*)(Wcol + k2 * P_STR + t * 32);
#pragma unroll
    for (int t = 0; t < 8; ++t) {
      acc[t] = __builtin_amdgcn_wmma_f32_16x16x4_f32(
          false, a, false, b[t], (short)0, acc[t], false, false);
    }
  }

#pragma unroll
  for (int t = 0; t < 8; ++t) {
    const int c = t * 16 + lr;
    const float bs = bias ? bias[c] : 0.0f;
#pragma unroll
    for (int i = 0; i < 8; ++i) {          // C/D layout: vgpr i -> M = i + 8*hi
      const int r = brow + mrow + hi * 8 + i;
      if (r < nrows) {
        float vv = acc[t][i] + bs;
        if (act) vv = geluf(vv);
        Cout[(size_t)r * DD + c] = vv;
      }
    }
  }
}

// ---------------------------------------------------------------------------
// GAT helpers
// ---------------------------------------------------------------------------
__global__ void nodeprep_kernel(const float* __restrict__ xh,
                                const float* __restrict__ a_s, const float* __restrict__ a_d,
                                float* __restrict__ als, float* __restrict__ ald) {
  const long long i = (long long)blockIdx.x * blockDim.x + threadIdx.x;
  if (i >= (long long)NN * HH) return;
  const int n = (int)(i >> 2), h = (int)(i & 3);
  const float* p  = xh + (size_t)n * DD + h * CC;
  const float* as = a_s + h * CC;
  const float* ad = a_d + h * CC;
  float s1 = 0.f, s2 = 0.f;
#pragma unroll
  for (int c = 0; c < CC; ++c) { const float x = p[c]; s1 += x * as[c]; s2 += x * ad[c]; }
  als[i] = s1; ald[i] = s2;
}

__global__ void fill_kernel(float* __restrict__ p, float v, long long n) {
  const long long i = (long long)blockIdx.x * blockDim.x + threadIdx.x;
  if (i < n) p[i] = v;
}

__global__ void axpby_kernel(float* __restrict__ out, const float* __restrict__ x, float a,
                             const float* __restrict__ y, float b, long long n) {
  const long long i = (long long)blockIdx.x * blockDim.x + threadIdx.x;
  if (i < n) out[i] = a * x[i] + b * y[i];
}

__global__ void edge_max_kernel(const long long* __restrict__ src, const long long* __restrict__ dst,
                                const float* __restrict__ als, const float* __restrict__ ald,
                                float* __restrict__ m) {
  const long long e = (long long)blockIdx.x * blockDim.x + threadIdx.x;
  if (e >= EE) return;
  const long long s = src[e], d = dst[e];
#pragma unroll
  for (int h = 0; h < HH; ++h) {
    float x = als[s * 4 + h] + ald[d * 4 + h];
    x = (x >= 0.f) ? x : 0.2f * x;  // leaky_relu(0.2)
    atomicMaxF(&m[d * 4 + h], x);
  }
}

__global__ void edge_exp_kernel(const long long* __restrict__ src, const long long* __restrict__ dst,
                                const float* __restrict__ als, const float* __restrict__ ald,
                                const float* __restrict__ m,
                                float* __restrict__ ebuf, float* __restrict__ ssum) {
  const long long e = (long long)blockIdx.x * blockDim.x + threadIdx.x;
  if (e >= EE) return;
  const long long s = src[e], d = dst[e];
#pragma unroll
  for (int h = 0; h < HH; ++h) {
    float x = als[s * 4 + h] + ald[d * 4 + h];
    x = (x >= 0.f) ? x : 0.2f * x;
    const float ev = expf(x - m[d * 4 + h]);
    ebuf[e * 4 + h] = ev;
    atomicAdd(&ssum[d * 4 + h], ev);
  }
}

// one wave per edge; lane = channel-within-head; coalesced gather + f32 atomic scatter
__global__ void edge_agg_kernel(const long long* __restrict__ src, const long long* __restrict__ dst,
                                const float* __restrict__ xh, const float* __restrict__ ebuf,
                                const float* __restrict__ ssum, float* __restrict__ agg) {
  const long long g = (long long)blockIdx.x * blockDim.x + threadIdx.x;
  const long long e = g >> 5;
  const int lane = (int)(g & 31);
  if (e >= EE) return;
  const long long s = src[e], d = dst[e];
  __builtin_prefetch(xh + (size_t)s * DD, 0, 0);
#pragma unroll
  for (int h = 0; h < HH; ++h) {
    const float alpha = ebuf[e * 4 + h] / (ssum[d * 4 + h] + 1e-16f);
    const int ch = h * CC + lane;
    atomicAdd(&agg[(size_t)d * DD + ch], xh[(size_t)s * DD + ch] * alpha);
  }
}

// fused: h = layer_norm(gelu(agg + bias)) ; one wave32 per node
__global__ void postnode_kernel(const float* __restrict__ agg, const float* __restrict__ bias,
                                const float* __restrict__ gg, const float* __restrict__ bb,
                                float* __restrict__ out) {
  const long long gid = (long long)blockIdx.x * blockDim.x + threadIdx.x;
  const int n = (int)(gid >> 5), lane = (int)(gid & 31);
  if (n >= NN) return;
  float vals[4]; float sum = 0.f;
#pragma unroll
  for (int j = 0; j < 4; ++j) {
    const int c = j * 32 + lane;
    float v = agg[(size_t)n * DD + c] + bias[c];
    v = geluf(v);
    vals[j] = v; sum += v;
  }
  for (int off = 16; off > 0; off >>= 1) sum += __shfl_xor(sum, off, 32);
  const float mean = sum * (1.0f / 128.0f);
  float var = 0.f;
#pragma unroll
  for (int j = 0; j < 4; ++j) { const float d = vals[j] - mean; var += d * d; }
  for (int off = 16; off > 0; off >>= 1) var += __shfl_xor(var, off, 32);
  var *= (1.0f / 128.0f);
  const float inv = rsqrtf(var + 1e-5f);
#pragma unroll
  for (int j = 0; j < 4; ++j) {
    const int c = j * 32 + lane;
    out[(size_t)n * DD + c] = (vals[j] - mean) * inv * gg[c] + bb[c];
  }
}

// ---------------------------------------------------------------------------
// Multihop broadcast
// ---------------------------------------------------------------------------
__global__ void edge_deg_kernel(const long long* __restrict__ dst, float* __restrict__ deg) {
  const long long e = (long long)blockIdx.x * blockDim.x + threadIdx.x;
  if (e >= EE) return;
  atomicAdd(&deg[dst[e]], 1.0f);
}

__global__ void edge_gather_kernel(const long long* __restrict__ src, const long long* __restrict__ dst,
                                   const float* __restrict__ cur, float* __restrict__ nxt) {
  const long long g = (long long)blockIdx.x * blockDim.x + threadIdx.x;
  const long long e = g >> 5;
  const int lane = (int)(g & 31);
  if (e >= EE) return;
  const long long s = src[e], d = dst[e];
#pragma unroll
  for (int j = 0; j < 4; ++j) {
    const int ch = j * 32 + lane;
    atomicAdd(&nxt[(size_t)d * DD + ch], cur[(size_t)s * DD + ch]);
  }
}

__global__ void hopscale_kernel(float* __restrict__ buf, const float* __restrict__ deg, long long n) {
  const long long i = (long long)blockIdx.x * blockDim.x + threadIdx.x;
  if (i < n) buf[i] /= fmaxf(deg[i >> 7], 1.0f);
}

// softmax over hops weighted combine; one wave32 per node
__global__ void final_kernel(const float* __restrict__ h0, const float* __restrict__ h1,
                             const float* __restrict__ h2, const float* __restrict__ h3,
                             const float* __restrict__ att, float* __restrict__ out) {
  const long long gid = (long long)blockIdx.x * blockDim.x + threadIdx.x;
  const int n = (int)(gid >> 5), lane = (int)(gid & 31);
  if (n >= NN) return;
  const float* hp[4] = {h0, h1, h2, h3};
  float al[4];
#pragma unroll
  for (int j = 0; j < 4; ++j) al[j] = att[j * 32 + lane];
  float hv[4][4];
  float sc[4];
#pragma unroll
  for (int k = 0; k < 4; ++k) {
    float p = 0.f;
#pragma unroll
    for (int j = 0; j < 4; ++j) {
      hv[k][j] = hp[k][(size_t)n * DD + j * 32 + lane];
      p += hv[k][j] * al[j];
    }
    for (int off = 16; off > 0; off >>= 1) p += __shfl_xor(p, off, 32);
    sc[k] = p;
  }
  const float mx = fmaxf(fmaxf(sc[0], sc[1]), fmaxf(sc[2], sc[3]));
  float w[4]; float sm = 0.f;
#pragma unroll
  for (int k = 0; k < 4; ++k) { w[k] = expf(sc[k] - mx); sm += w[k]; }
  const float invs = 1.0f / sm;
#pragma unroll
  for (int j = 0; j < 4; ++j) {
    float o = 0.f;
#pragma unroll
    for (int k = 0; k < 4; ++k) o += w[k] * invs * hv[k][j];
    out[(size_t)n * DD + j * 32 + lane] = o;
  }
}

// ---------------------------------------------------------------------------
// Launch
// ---------------------------------------------------------------------------
extern "C" void kernel_launch(void* const* d_in, const int* in_sizes, int n_in,
                              void* d_out, int out_size, void* d_ws, size_t ws_size,
                              hipStream_t stream) {
  (void)in_sizes; (void)n_in; (void)out_size; (void)ws_size;
  const float* x      = (const float*)d_in[0];
  const long long* ei = (const long long*)d_in[1];
  const long long* src = ei;
  const long long* dst = ei + EE;
  const float* W_in  = (const float*)d_in[2];
  const float* b_in  = (const float*)d_in[3];
  const float* W1    = (const float*)d_in[4];
  const float* a_s1  = (const float*)d_in[5];
  const float* a_d1  = (const float*)d_in[6];
  const float* b1    = (const float*)d_in[7];
  const float* W2    = (const float*)d_in[8];
  const float* a_s2  = (const float*)d_in[9];
  const float* a_d2  = (const float*)d_in[10];
  const float* b2    = (const float*)d_in[11];
  const float* g1    = (const float*)d_in[12];
  const float* be1   = (const float*)d_in[13];
  const float* g2    = (const float*)d_in[14];
  const float* be2   = (const float*)d_in[15];
  const float* W_ode = (const float*)d_in[16];
  const float* b_ode = (const float*)d_in[17];
  const float* att   = (const float*)d_in[18];
  float* out = (float*)d_out;

  // workspace layout (~132 MB of f32)
  float* ws = (float*)d_ws;
  const size_t NB = (size_t)NN * DD;
  float* Wn_in  = ws;                  // 16384
  float* Wn_ode = ws + 16384;          // 16384
  float* B0 = ws + 32768;              // node features / hop0
  float* B1 = B0 + NB;                 // xh / ODE tmp / hop1
  float* B2 = B1 + NB;                 // GAT agg / ODE k / hop2
  float* B3 = B2 + NB;                 // ODE acc / hop3
  float* als  = B3 + NB;               // N*4
  float* ald  = als + (size_t)NN * HH;
  float* mbuf = ald + (size_t)NN * HH;
  float* sbuf = mbuf + (size_t)NN * HH;
  float* deg  = sbuf + (size_t)NN * HH;  // N
  float* ebuf = deg + NN;                // E*4

  const int TB = 256;
  const int gemm_blocks = (NN + 127) / 128;
  auto cdiv = [](long long a, long long b) { return (int)((a + b - 1) / b); };
  const long long nNH = (long long)NN * HH;
  const long long nNB = (long long)NB;
  const long long nEW = EE * 32;  // wave-per-edge threads
  const long long nNW = (long long)NN * 32;

  // 1) spectral norms for W_in / W_ode
  spectral_kernel<<<2, 128, 0, stream>>>(W_in, W_ode, Wn_in, Wn_ode);

  // 2) h = gelu(x @ Wn_in + b_in)
  gemm128_kernel<<<gemm_blocks, TB, GEMM_LDS_BYTES, stream>>>(x, Wn_in, b_in, B0, NN, 1);

  // 3) two GAT layers
  auto gat = [&](const float* W, const float* as_, const float* ad_, const float* bias,
                 const float* gg, const float* bb) {
    gemm128_kernel<<<gemm_blocks, TB, GEMM_LDS_BYTES, stream>>>(B0, W, nullptr, B1, NN, 0);
    nodeprep_kernel<<<cdiv(nNH, TB), TB, 0, stream>>>(B1, as_, ad_, als, ald);
    fill_kernel<<<cdiv(nNH, TB), TB, 0, stream>>>(mbuf, -INFINITY, nNH);
    fill_kernel<<<cdiv(nNH, TB), TB, 0, stream>>>(sbuf, 0.0f, nNH);
    fill_kernel<<<cdiv(nNB, TB), TB, 0, stream>>>(B2, 0.0f, nNB);
    edge_max_kernel<<<cdiv(EE, TB), TB, 0, stream>>>(src, dst, als, ald, mbuf);
    edge_exp_kernel<<<cdiv(EE, TB), TB, 0, stream>>>(src, dst, als, ald, mbuf, ebuf, sbuf);
    edge_agg_kernel<<<cdiv(nEW, TB), TB, 0, stream>>>(src, dst, B1, ebuf, sbuf, B2);
    postnode_kernel<<<cdiv(nNW, TB), TB, 0, stream>>>(B2, bias, gg, bb, B0);
  };
  gat(W1, a_s1, a_d1, b1, g1, be1);
  gat(W2, a_s2, a_d2, b2, g2, be2);

  // 4) ODE block: RK4, 4 steps, f(y) = gelu(y @ Wn_ode + b_ode)
  const float dtt = 0.25f;
  for (int s6 = 0; s6 < 4; ++s6) {
    gemm128_kernel<<<gemm_blocks, TB, GEMM_LDS_BYTES, stream>>>(B0, Wn_ode, b_ode, B2, NN, 1);   // k1
    axpby_kernel<<<cdiv(nNB, TB), TB, 0, stream>>>(B3, B2, 1.0f, B2, 0.0f, nNB);                 // acc=k1
    axpby_kernel<<<cdiv(nNB, TB), TB, 0, stream>>>(B1, B0, 1.0f, B2, 0.5f * dtt, nNB);           // tmp
    gemm128_kernel<<<gemm_blocks, TB, GEMM_LDS_BYTES, stream>>>(B1, Wn_ode, b_ode, B2, NN, 1);   // k2
    axpby_kernel<<<cdiv(nNB, TB), TB, 0, stream>>>(B3, B3, 1.0f, B2, 2.0f, nNB);
    axpby_kernel<<<cdiv(nNB, TB), TB, 0, stream>>>(B1, B0, 1.0f, B2, 0.5f * dtt, nNB);
    gemm128_kernel<<<gemm_blocks, TB, GEMM_LDS_BYTES, stream>>>(B1, Wn_ode, b_ode, B2, NN, 1);   // k3
    axpby_kernel<<<cdiv(nNB, TB), TB, 0, stream>>>(B3, B3, 1.0f, B2, 2.0f, nNB);
    axpby_kernel<<<cdiv(nNB, TB), TB, 0, stream>>>(B1, B0, 1.0f, B2, dtt, nNB);
    gemm128_kernel<<<gemm_blocks, TB, GEMM_LDS_BYTES, stream>>>(B1, Wn_ode, b_ode, B2, NN, 1);   // k4
    axpby_kernel<<<cdiv(nNB, TB), TB, 0, stream>>>(B3, B3, 1.0f, B2, 1.0f, nNB);
    axpby_kernel<<<cdiv(nNB, TB), TB, 0, stream>>>(B0, B0, 1.0f, B3, dtt / 6.0f, nNB);           // y+=dt/6*acc
  }

  // 5) multihop broadcast (3 hops of mean aggregation) + softmax combine
  fill_kernel<<<cdiv((long long)NN, TB), TB, 0, stream>>>(deg, 0.0f, (long long)NN);
  edge_deg_kernel<<<cdiv(EE, TB), TB, 0, stream>>>(dst, deg);
  float* hops[4] = {B0, B1, B2, B3};
  for (int k = 1; k <= 3; ++k) {
    fill_kernel<<<cdiv(nNB, TB), TB, 0, stream>>>(hops[k], 0.0f, nNB);
    edge_gather_kernel<<<cdiv(nEW, TB), TB, 0, stream>>>(src, dst, hops[k - 1], hops[k]);
    hopscale_kernel<<<cdiv(nNB, TB), TB, 0, stream>>>(hops[k], deg, nNB);
  }
  final_kernel<<<cdiv(nNW, TB), TB, 0, stream>>>(B0, B1, B2, B3, att, out);
}